// SummaryStats_34282428957035
// MI455X (gfx1250) — compile-verified
//
#include <hip/hip_runtime.h>
#include <cstdint>
#include <cstddef>

#define NBINS 31
#define BS 256
#define HREP 16   // histogram replicas: one per half-wave (BS/32 waves * 2)

// ---------------------------------------------------------------------------
// CDNA5 async global->LDS path (gfx1250), signature confirmed by diagnostics:
// (v4i AS1*, v4i AS3*, imm offset, imm cpol); s_wait_asynccnt takes an imm.
// ---------------------------------------------------------------------------
#if defined(__HIP_DEVICE_COMPILE__) && \
    __has_builtin(__builtin_amdgcn_global_load_async_to_lds_b128) && \
    __has_builtin(__builtin_amdgcn_s_wait_asynccnt)
#define USE_ASYNC 1
#else
#define USE_ASYNC 0
#endif

#if USE_ASYNC
typedef int v4i __attribute__((ext_vector_type(4)));
typedef __attribute__((address_space(1))) v4i* gp_t;   // global int4*
typedef __attribute__((address_space(3))) v4i* lp_t;   // LDS int4*

__device__ __forceinline__ void async_copy16(const void* gsrc, void* ldst) {
  __builtin_amdgcn_global_load_async_to_lds_b128(
      (gp_t)(unsigned long long)gsrc,
      (lp_t)(unsigned int)(unsigned long long)ldst,
      0, 0);
}
#endif

// Order-preserving float -> uint32 map (for atomic min/max on floats).
__device__ __forceinline__ unsigned f2o(float f) {
  unsigned u = __float_as_uint(f);
  return (u & 0x80000000u) ? ~u : (u | 0x80000000u);
}
__device__ __forceinline__ float o2f(unsigned o) {
  unsigned u = (o & 0x80000000u) ? (o ^ 0x80000000u) : ~o;
  return __uint_as_float(u);
}

// ---------------------------------------------------------------------------
// Kernels.  "Supertile" = BS*2 float4 = 2048 floats per block per step.
// ---------------------------------------------------------------------------
__global__ void ss_init(float* __restrict__ out, int out_n,
                        unsigned* __restrict__ wsU, float* __restrict__ wsF) {
  for (int i = threadIdx.x; i < out_n; i += blockDim.x) out[i] = 0.0f;
  if (threadIdx.x == 0) {
    wsU[0] = 0xFFFFFFFFu;  // min sentinel
    wsU[1] = 0x00000000u;  // max sentinel
    wsU[2] = 0u;           // nonzero count
    wsF[4] = 0.0f;         // sum
    wsF[5] = 0.0f;         // sum of squares
  }
}

__global__ __launch_bounds__(BS) void ss_pass1(const float* __restrict__ x,
                                               long long n, int ntiles,
                                               unsigned* __restrict__ wsU,
                                               float* __restrict__ wsF) {
  const int tid = threadIdx.x;
  const float4* __restrict__ x4 = (const float4*)x;

  float lmin = __builtin_inff(), lmax = -__builtin_inff();
  float lsum = 0.0f, lss = 0.0f;
  unsigned lnz = 0;

  auto acc = [&](float v) {
    lmin = fminf(lmin, v);
    lmax = fmaxf(lmax, v);
    lsum += v;
    lss = fmaf(v, v, lss);
    lnz += (v != 0.0f) ? 1u : 0u;
  };

#if USE_ASYNC
  __shared__ float4 buf[2][2][BS];
  auto issue = [&](int t, int s) {
    const size_t base = (size_t)t * (BS * 2) + tid;
    async_copy16(&x4[base],      &buf[s][0][tid]);
    async_copy16(&x4[base + BS], &buf[s][1][tid]);
  };
  int t = (int)blockIdx.x;
  int stage = 0;
  if (t < ntiles) issue(t, 0);                   // forward walk
  while (t < ntiles) {
    int tn = t + (int)gridDim.x;
    if (tn < ntiles) {
      issue(tn, stage ^ 1);
      __builtin_amdgcn_s_wait_asynccnt(2);       // current stage landed
    } else {
      __builtin_amdgcn_s_wait_asynccnt(0);
    }
    float4 a = buf[stage][0][tid];
    float4 b = buf[stage][1][tid];
    acc(a.x); acc(a.y); acc(a.z); acc(a.w);
    acc(b.x); acc(b.y); acc(b.z); acc(b.w);
    stage ^= 1;
    t = tn;
  }
#else
  for (int t = (int)blockIdx.x; t < ntiles; t += (int)gridDim.x) {
    const size_t base = (size_t)t * (BS * 2) + tid;
    float4 a = x4[base];
    float4 b = x4[base + BS];
    acc(a.x); acc(a.y); acc(a.z); acc(a.w);
    acc(b.x); acc(b.y); acc(b.z); acc(b.w);
  }
#endif
  // scalar remainder (n not a multiple of 2048)
  for (long long i = (long long)ntiles * (BS * 8) + (long long)blockIdx.x * BS + tid;
       i < n; i += (long long)gridDim.x * BS)
    acc(x[i]);

  // wave32 reduction
  for (int off = 16; off > 0; off >>= 1) {
    lmin = fminf(lmin, __shfl_xor(lmin, off));
    lmax = fmaxf(lmax, __shfl_xor(lmax, off));
    lsum += __shfl_xor(lsum, off);
    lss += __shfl_xor(lss, off);
    lnz += (unsigned)__shfl_xor((int)lnz, off);
  }

  __shared__ float rmin[BS / 32], rmax[BS / 32], rsum[BS / 32], rss[BS / 32];
  __shared__ unsigned rnz[BS / 32];
  const int w = tid >> 5;
  if ((tid & 31) == 0) { rmin[w] = lmin; rmax[w] = lmax; rsum[w] = lsum; rss[w] = lss; rnz[w] = lnz; }
  __syncthreads();
  if (tid == 0) {
    float bmin = rmin[0], bmax = rmax[0], bsum = rsum[0], bss = rss[0];
    unsigned bnz = rnz[0];
    for (int i = 1; i < BS / 32; ++i) {
      bmin = fminf(bmin, rmin[i]);
      bmax = fmaxf(bmax, rmax[i]);
      bsum += rsum[i];
      bss += rss[i];
      bnz += rnz[i];
    }
    atomicMin(&wsU[0], f2o(bmin));
    atomicMax(&wsU[1], f2o(bmax));
    atomicAdd(&wsU[2], bnz);
    atomicAdd(&wsF[4], bsum);
    atomicAdd(&wsF[5], bss);
  }
}

__global__ void ss_finalize(const unsigned* __restrict__ wsU,
                            const float* __restrict__ wsF,
                            float* __restrict__ out, float num) {
  const int tid = threadIdx.x;
  const float mn = o2f(wsU[0]);
  const float mx = o2f(wsU[1]);
  if (tid == 0) {
    out[0] = mn;
    out[1] = mx;
    out[2] = num;
    out[3] = (float)wsU[2];
    out[4] = wsF[4];
    out[5] = wsF[5];
  }
  const float step = (mx - mn) / (float)NBINS;
  if (tid < NBINS + 1)
    out[6 + NBINS + tid] = (tid == NBINS) ? mx : mn + step * (float)tid;
}

__global__ __launch_bounds__(BS) void ss_pass2(const float* __restrict__ x,
                                               long long n, int ntiles,
                                               float* __restrict__ out) {
  const int tid = threadIdx.x;
  const float4* __restrict__ x4 = (const float4*)x;

  __shared__ int hist[HREP][32];       // half-wave-private replicas
  __shared__ float edges[NBINS + 1];
  for (int i = tid; i < HREP * 32; i += BS) ((int*)hist)[i] = 0;

  const float mn = out[0];
  const float mx = out[1];
  const float step = (mx - mn) / (float)NBINS;
  if (tid < NBINS + 1) edges[tid] = (tid == NBINS) ? mx : mn + step * (float)tid;
  __syncthreads();

  const float scale = (mx > mn) ? (float)NBINS / (mx - mn) : 0.0f;
  const int rep = tid >> 4;            // 16 lanes per replica

  auto binit = [&](float v) {
    int b = (int)((v - mn) * scale);
    b = b < 0 ? 0 : (b > NBINS - 1 ? NBINS - 1 : b);
    // match searchsorted(edges, v, 'right')-1 semantics at bin boundaries
    if (b < NBINS - 1 && v >= edges[b + 1]) ++b;
    else if (b > 0 && v < edges[b]) --b;
    atomicAdd(&hist[rep][b], 1);
  };

#if USE_ASYNC
  __shared__ float4 buf[2][2][BS];
  auto issue = [&](int t, int s) {
    const size_t base = (size_t)t * (BS * 2) + tid;
    async_copy16(&x4[base],      &buf[s][0][tid]);
    async_copy16(&x4[base + BS], &buf[s][1][tid]);
  };
  // REVERSE tile walk: first touch what pass1 read last (still in 192MB L2).
  int t = ((int)blockIdx.x < ntiles)
              ? (int)blockIdx.x + (int)gridDim.x * ((ntiles - 1 - (int)blockIdx.x) / (int)gridDim.x)
              : -1;
  int stage = 0;
  if (t >= 0) issue(t, 0);
  while (t >= 0) {
    int tn = t - (int)gridDim.x;
    if (tn >= 0) {
      issue(tn, stage ^ 1);
      __builtin_amdgcn_s_wait_asynccnt(2);
    } else {
      __builtin_amdgcn_s_wait_asynccnt(0);
    }
    float4 a = buf[stage][0][tid];
    float4 b = buf[stage][1][tid];
    binit(a.x); binit(a.y); binit(a.z); binit(a.w);
    binit(b.x); binit(b.y); binit(b.z); binit(b.w);
    stage ^= 1;
    t = tn;
  }
#else
  for (int t = (int)blockIdx.x; t < ntiles; t += (int)gridDim.x) {
    const size_t base = (size_t)t * (BS * 2) + tid;
    float4 a = x4[base];
    float4 b = x4[base + BS];
    binit(a.x); binit(a.y); binit(a.z); binit(a.w);
    binit(b.x); binit(b.y); binit(b.z); binit(b.w);
  }
#endif
  for (long long i = (long long)ntiles * (BS * 8) + (long long)blockIdx.x * BS + tid;
       i < n; i += (long long)gridDim.x * BS)
    binit(x[i]);

  __syncthreads();
  if (tid < NBINS) {
    int c = 0;
    for (int r = 0; r < HREP; ++r) c += hist[r][tid];
    if (c) atomicAdd(&out[6 + tid], (float)c);
  }
}

// ---------------------------------------------------------------------------
// Launch
// ---------------------------------------------------------------------------
extern "C" void kernel_launch(void* const* d_in, const int* in_sizes, int n_in,
                              void* d_out, int out_size, void* d_ws, size_t ws_size,
                              hipStream_t stream) {
  const float* x = (const float*)d_in[0];
  const long long n = (long long)in_sizes[0];
  float* out = (float*)d_out;
  unsigned* wsU = (unsigned*)d_ws;
  float* wsF = (float*)d_ws;

  const int ntiles = (int)(n / (8 * BS));   // supertile = 2048 floats
  int grid = 2048;
  if (ntiles > 0 && grid > ntiles) grid = ntiles;
  if (ntiles == 0) grid = 1;

  ss_init<<<1, 256, 0, stream>>>(out, out_size, wsU, wsF);
  ss_pass1<<<grid, BS, 0, stream>>>(x, n, ntiles, wsU, wsF);
  ss_finalize<<<1, 64, 0, stream>>>(wsU, wsF, out, (float)n);
  ss_pass2<<<grid, BS, 0, stream>>>(x, n, ntiles, out);
}